// DifferentiableKMeans_77043123356280
// MI455X (gfx1250) — compile-verified
//
#include <hip/hip_runtime.h>

typedef __attribute__((ext_vector_type(2))) float v2f;
typedef __attribute__((ext_vector_type(8))) float v8f;

#define N_POINTS  32768
#define N_CENTERS 512
#define DIM       128
#define TOPK      10
#define LDS_STRIDE (N_CENTERS + 4)   // pad: 8*stride % 64 != 0 -> no bank clash between lane halves

// ---- Kernel 1: per-center squared norms into workspace ----
__global__ void c2_kernel(const float* __restrict__ centers, float* __restrict__ c2) {
    int c = blockIdx.x * blockDim.x + threadIdx.x;
    if (c < N_CENTERS) {
        const float4* row = (const float4*)(centers + (size_t)c * DIM);
        float s = 0.f;
#pragma unroll
        for (int i = 0; i < DIM / 4; ++i) {
            float4 v = row[i];
            s += v.x * v.x + v.y * v.y + v.z * v.z + v.w * v.w;
        }
        c2[c] = s;
    }
}

// ---- Kernel 2: WMMA scores -> per-row top-10 -> gather x rows ----
__launch_bounds__(512, 1)
__global__ void kmeans_topk_gather(const float* __restrict__ x,
                                   const float* __restrict__ centers,
                                   const float* __restrict__ c2,
                                   float* __restrict__ out) {
    __shared__ float dist[16 * LDS_STRIDE];

    const int tid   = threadIdx.x;
    const int lane  = tid & 31;
    const int wave  = tid >> 5;          // 0..15
    const int mbase = blockIdx.x * 16;   // 16 x-rows per block

    // WMMA f32 16x16x4 A/B lane layout: lanes 0-15 hold K={0,1}, lanes 16-31 hold K={2,3}
    const int mrow = lane & 15;
    const int koff = (lane >> 4) * 2;

    // --- Phase 1: score s[m][n] = c2[n] - 2*(x_m . c_n); x^2 and sqrt are
    // monotone per-row so they never change the top-k ordering. ---
#pragma unroll
    for (int t = 0; t < 2; ++t) {
        const int nbase = (wave + t * 16) * 16;          // 32 n-tiles over 16 waves
        v8f acc = {};
        const float* aptr = x       + (size_t)(mbase + mrow) * DIM + koff;
        const float* bptr = centers + (size_t)(nbase + mrow) * DIM + koff;
#pragma unroll
        for (int k0 = 0; k0 < DIM; k0 += 4) {
            v2f a = *(const v2f*)(aptr + k0);
            v2f b = *(const v2f*)(bptr + k0);
            // D = A(16x4) * B(4x16) + C, f32 in/out
            acc = __builtin_amdgcn_wmma_f32_16x16x4_f32(false, a, false, b,
                                                        (short)0, acc, false, false);
        }
        // C/D layout: VGPR r, lanes 0-15 -> M=r, N=lane; lanes 16-31 -> M=r+8, N=lane-16
        const int nn   = nbase + (lane & 15);
        const float cn = c2[nn];
        const int mofs = (lane >> 4) * 8;
#pragma unroll
        for (int r = 0; r < 8; ++r) {
            dist[(r + mofs) * LDS_STRIDE + nn] = cn - 2.0f * acc[r];
        }
    }
    __syncthreads();

    // --- Phase 2: wave w owns row w: 10x (min-reduce over 512, mark, gather) ---
    float* drow = dist + wave * LDS_STRIDE;
    const size_t grow = (size_t)(mbase + wave);
    const float4* x4  = (const float4*)x;
    float4*       o4  = (float4*)out;

    for (int j = 0; j < TOPK; ++j) {
        float vmin = 3.4e38f;
        int   imin = N_CENTERS;
#pragma unroll
        for (int i = lane; i < N_CENTERS; i += 32) {
            float v = drow[i];
            if (v < vmin || (v == vmin && i < imin)) { vmin = v; imin = i; }
        }
#pragma unroll
        for (int off = 16; off > 0; off >>= 1) {
            float ov = __shfl_xor(vmin, off, 32);
            int   oi = __shfl_xor(imin, off, 32);
            if (ov < vmin || (ov == vmin && oi < imin)) { vmin = ov; imin = oi; }
        }
        if (lane == 0) drow[imin] = 3.4e38f;   // exclude winner (same-wave LDS dep)

        // gather x[imin]: 128 f32 = 32 lanes x float4 (row cached in L2, <=512 distinct)
        o4[(grow * TOPK + j) * (DIM / 4) + lane] = x4[(size_t)imin * (DIM / 4) + lane];
    }
}

extern "C" void kernel_launch(void* const* d_in, const int* in_sizes, int n_in,
                              void* d_out, int out_size, void* d_ws, size_t ws_size,
                              hipStream_t stream) {
    (void)in_sizes; (void)n_in; (void)out_size; (void)ws_size;
    const float* x       = (const float*)d_in[0];
    const float* centers = (const float*)d_in[1];
    float* out = (float*)d_out;
    float* c2  = (float*)d_ws;   // 512 floats of scratch

    c2_kernel<<<(N_CENTERS + 255) / 256, 256, 0, stream>>>(centers, c2);
    kmeans_topk_gather<<<N_POINTS / 16, 512, 0, stream>>>(x, centers, c2, out);
}